// DynamicHypergraphLayer_50096498540669
// MI455X (gfx1250) — compile-verified
//
#include <hip/hip_runtime.h>
#include <cstdint>

#define N_NODES 20000
#define N_EDGES 8192
#define F_DIM   128

typedef float v2f __attribute__((ext_vector_type(2)));
typedef float v8f __attribute__((ext_vector_type(8)));

// Lower 32 bits of a flat shared-aperture address are the LDS byte offset (ISA 10.2).
__device__ __forceinline__ uint32_t lds_off(const void* p) {
  return (uint32_t)(uintptr_t)p;
}

// CDNA5 async copies: global -> LDS per lane, tracked by ASYNCcnt.
__device__ __forceinline__ void async_copy16(uint32_t lds_dst, const void* src) {
  asm volatile("global_load_async_to_lds_b128 %0, %1, off"
               :: "v"(lds_dst), "v"((uint64_t)(uintptr_t)src)
               : "memory");
}
__device__ __forceinline__ void async_copy8(uint32_t lds_dst, const void* src) {
  asm volatile("global_load_async_to_lds_b64 %0, %1, off"
               :: "v"(lds_dst), "v"((uint64_t)(uintptr_t)src)
               : "memory");
}

__device__ __forceinline__ float gelu_erf(float v) {
  return 0.5f * v * (1.0f + erff(v * 0.70710678118654752440f));
}

// ---------------------------------------------------------------- small kernels

__global__ __launch_bounds__(256) void hg_zero(float* __restrict__ De) {
  De[blockIdx.x * 256 + threadIdx.x] = 0.0f;
}

// Per-row: Dv, dv^-1/2, and scaled linear projection xs = dv^-1/2 * (x @ lin_w^T + b)
__global__ __launch_bounds__(256) void hg_row(
    const float* __restrict__ x, const float* __restrict__ H,
    const float* __restrict__ lw, const float* __restrict__ lb,
    float* __restrict__ dvis, float* __restrict__ xs) {
  __shared__ float red[8];
  __shared__ float xrow[F_DIM];
  __shared__ float part[256];
  const int i = blockIdx.x;
  const int t = threadIdx.x;

  const float* hr = H + (size_t)i * N_EDGES;
  float s = 0.0f;
#pragma unroll
  for (int p = 0; p < 8; ++p) {
    const float4 v = *(const float4*)(hr + 4 * (t + 256 * p));
    s += v.x + v.y + v.z + v.w;
  }
#pragma unroll
  for (int off = 16; off > 0; off >>= 1) s += __shfl_xor(s, off);
  if ((t & 31) == 0) red[t >> 5] = s;
  if (t < F_DIM) xrow[t] = x[(size_t)i * F_DIM + t];
  __syncthreads();
  if (t == 0) {
    float tot = 0.0f;
#pragma unroll
    for (int w = 0; w < 8; ++w) tot += red[w];
    const float d = rsqrtf(fmaxf(tot, 1.0f));
    dvis[i] = d;
    red[0] = d;
  }
  __syncthreads();
  const float d = red[0];

  const int f = t & 127;
  const int h = t >> 7;
  const float* wr = lw + (size_t)f * F_DIM + h * 64;
  const float* xr = xrow + h * 64;
  float acc = 0.0f;
#pragma unroll 8
  for (int k = 0; k < 64; ++k) acc = fmaf(wr[k], xr[k], acc);
  part[t] = acc;
  __syncthreads();
  if (t < F_DIM)
    xs[(size_t)i * F_DIM + t] = (part[t] + part[t + 128] + lb[t]) * d;
}

// Column sums of H (hyperedge degrees), chunked over rows with atomic combine.
__global__ __launch_bounds__(256) void hg_col(const float* __restrict__ H,
                                              float* __restrict__ De) {
  const int col = blockIdx.x * 256 + threadIdx.x;
  const size_t r0 = (size_t)blockIdx.y * 500;
  const float* p = H + r0 * N_EDGES + col;
  float s = 0.0f;
#pragma unroll 4
  for (int r = 0; r < 500; ++r) s += p[(size_t)r * N_EDGES];
  atomicAdd(&De[col], s);
}

__global__ __launch_bounds__(256) void hg_wde(const float* __restrict__ W,
                                              const float* __restrict__ De,
                                              float* __restrict__ wde) {
  const int e = blockIdx.x * 256 + threadIdx.x;
  wde[e] = W[e] / fmaxf(De[e], 1.0f);
}

// ---------------------------------------------------------------- GEMM 1
// ts[e,f] = wde[e] * sum_k H[k,e] * xs[k,f]     (A = H^T, K = N_NODES)
// Block tile: M=32 (edges) x N=128 (features); 8 waves, each 16x32 (2 accums).
// Double-buffered LDS stages fed by async copies (1x b64 + 2x b128 per thread).
__global__ __launch_bounds__(256) void hg_gemm1(
    const float* __restrict__ H, const float* __restrict__ xs,
    const float* __restrict__ wde, float* __restrict__ ts) {
  __shared__ float As[2][16][48];    // [buf][k][e_local], stride 48 -> conflict-free
  __shared__ float Bs[2][16][128];   // [buf][k][f]

  const int t = threadIdx.x;
  const int wave = t >> 5, lane = t & 31;
  const int mt = wave & 1, nh = wave >> 1;
  const int kh = lane >> 4, nl = lane & 15;
  const int e0 = blockIdx.x * 32;

  const int kA = t >> 4, cA = t & 15;              // A tile: 16 rows x 16 b64 chunks
  const uint32_t aBuf = 16 * 48 * 4;
  const uint32_t bBuf = 16 * 128 * 4;
  const uint32_t a_dst = lds_off(&As[0][kA][cA * 2]);
  const uint32_t b_dst = lds_off(&Bs[0][0][0]) + t * 16;
  const float* aSrc = H + (size_t)kA * N_EDGES + e0 + cA * 2;

  v8f acc[2];
#pragma unroll
  for (int j = 0; j < 2; ++j) acc[j] = (v8f){0, 0, 0, 0, 0, 0, 0, 0};

  // prologue: stage 0 into buffer 0
  async_copy8(a_dst, aSrc);
  async_copy16(b_dst, xs + t * 4);
  async_copy16(b_dst + 4096, xs + (t + 256) * 4);

  const int S = N_NODES / 16;
  for (int s = 0; s < S; ++s) {
    const int cur = s & 1;
    if (s + 1 < S) {                               // prefetch next stage
      const size_t k1 = (size_t)(s + 1) * 16;
      const uint32_t nb = (uint32_t)(cur ^ 1);
      async_copy8(a_dst + nb * aBuf, aSrc + k1 * N_EDGES);
      async_copy16(b_dst + nb * bBuf, xs + k1 * F_DIM + t * 4);
      async_copy16(b_dst + nb * bBuf + 4096, xs + k1 * F_DIM + (t + 256) * 4);
      __builtin_amdgcn_s_wait_asynccnt(3);         // oldest 3 (this stage) done
    } else {
      __builtin_amdgcn_s_wait_asynccnt(0);
    }
    __syncthreads();
#pragma unroll
    for (int ks = 0; ks < 4; ++ks) {
      const int r = ks * 4 + kh * 2;
      v2f a;
      a.x = As[cur][r][mt * 16 + nl];
      a.y = As[cur][r + 1][mt * 16 + nl];
#pragma unroll
      for (int j = 0; j < 2; ++j) {
        const int n = nh * 32 + j * 16 + nl;
        v2f b;
        b.x = Bs[cur][r][n];
        b.y = Bs[cur][r + 1][n];
        acc[j] = __builtin_amdgcn_wmma_f32_16x16x4_f32(
            false, a, false, b, (short)0, acc[j], false, false);
      }
    }
    __syncthreads();
  }

  float w8[8];
#pragma unroll
  for (int v = 0; v < 8; ++v) w8[v] = wde[e0 + mt * 16 + kh * 8 + v];
#pragma unroll
  for (int j = 0; j < 2; ++j) {
#pragma unroll
    for (int v = 0; v < 8; ++v) {
      const int e = e0 + mt * 16 + kh * 8 + v;
      ts[(size_t)e * F_DIM + nh * 32 + j * 16 + nl] = acc[j][v] * w8[v];
    }
  }
}

// ---------------------------------------------------------------- GEMM 2
// out[i,f] = gelu( dvis[i] * sum_e H[i,e] * ts[e,f] )   (K = N_EDGES)
// Block tile: M=64 x N=128; 8 waves, each 16x64 (4 accums); double-buffered.
__global__ __launch_bounds__(256) void hg_gemm2(
    const float* __restrict__ H, const float* __restrict__ ts,
    const float* __restrict__ dvis, float* __restrict__ out) {
  __shared__ float As[2][64][20];    // [buf][m][k], stride 20 -> conflict-free v2f
  __shared__ float Bs[2][16][128];

  const int t = threadIdx.x;
  const int wave = t >> 5, lane = t & 31;
  const int mt = wave & 3, nh = wave >> 2;
  const int kh = lane >> 4, nl = lane & 15;
  const int i0 = blockIdx.x * 64;

  const int mA = t >> 2, cAq = t & 3;              // A tile: 64 rows x 4 b128 chunks
  int rowA = i0 + mA;
  if (rowA > N_NODES - 1) rowA = N_NODES - 1;      // clamp tail loads
  const uint32_t aBuf = 64 * 20 * 4;
  const uint32_t bBuf = 16 * 128 * 4;
  const uint32_t a_dst = lds_off(&As[0][mA][cAq * 4]);
  const uint32_t b_dst = lds_off(&Bs[0][0][0]) + t * 16;
  const float* aSrc = H + (size_t)rowA * N_EDGES + cAq * 4;

  v8f acc[4];
#pragma unroll
  for (int j = 0; j < 4; ++j) acc[j] = (v8f){0, 0, 0, 0, 0, 0, 0, 0};

  // prologue: stage 0 into buffer 0
  async_copy16(a_dst, aSrc);
  async_copy16(b_dst, ts + t * 4);
  async_copy16(b_dst + 4096, ts + (t + 256) * 4);

  const int S = N_EDGES / 16;
  for (int s = 0; s < S; ++s) {
    const int cur = s & 1;
    if (s + 1 < S) {
      const size_t k1 = (size_t)(s + 1) * 16;
      const uint32_t nb = (uint32_t)(cur ^ 1);
      async_copy16(a_dst + nb * aBuf, aSrc + k1);
      async_copy16(b_dst + nb * bBuf, ts + k1 * F_DIM + t * 4);
      async_copy16(b_dst + nb * bBuf + 4096, ts + k1 * F_DIM + (t + 256) * 4);
      __builtin_amdgcn_s_wait_asynccnt(3);
    } else {
      __builtin_amdgcn_s_wait_asynccnt(0);
    }
    __syncthreads();
#pragma unroll
    for (int ks = 0; ks < 4; ++ks) {
      const int r = ks * 4 + kh * 2;
      const v2f a = *(const v2f*)&As[cur][mt * 16 + nl][r];  // 8B-aligned
#pragma unroll
      for (int j = 0; j < 4; ++j) {
        const int n = nh * 64 + j * 16 + nl;
        v2f b;
        b.x = Bs[cur][r][n];
        b.y = Bs[cur][r + 1][n];
        acc[j] = __builtin_amdgcn_wmma_f32_16x16x4_f32(
            false, a, false, b, (short)0, acc[j], false, false);
      }
    }
    __syncthreads();
  }

  float d8[8];
#pragma unroll
  for (int v = 0; v < 8; ++v) {
    int i = i0 + mt * 16 + kh * 8 + v;
    if (i > N_NODES - 1) i = N_NODES - 1;
    d8[v] = dvis[i];
  }
#pragma unroll
  for (int j = 0; j < 4; ++j) {
#pragma unroll
    for (int v = 0; v < 8; ++v) {
      const int i = i0 + mt * 16 + kh * 8 + v;
      if (i < N_NODES)
        out[(size_t)i * F_DIM + nh * 64 + j * 16 + nl] =
            gelu_erf(acc[j][v] * d8[v]);
    }
  }
}

// ---------------------------------------------------------------- launcher

extern "C" void kernel_launch(void* const* d_in, const int* in_sizes, int n_in,
                              void* d_out, int out_size, void* d_ws, size_t ws_size,
                              hipStream_t stream) {
  const float* x     = (const float*)d_in[0];   // [N, 128]
  const float* H     = (const float*)d_in[1];   // [N, E]
  const float* W     = (const float*)d_in[2];   // [E]
  const float* lin_w = (const float*)d_in[3];   // [128, 128]
  const float* lin_b = (const float*)d_in[4];   // [128]
  float* out = (float*)d_out;
  float* ws  = (float*)d_ws;

  float* dvis = ws;                                  // 20000
  float* De   = ws + 20000;                          // 8192
  float* wde  = ws + 28192;                          // 8192
  float* xs   = ws + 36384;                          // 20000*128 (16B aligned)
  float* ts   = ws + 36384 + (size_t)N_NODES * F_DIM;// 8192*128  (16B aligned)

  hg_zero<<<N_EDGES / 256, 256, 0, stream>>>(De);
  hg_row<<<N_NODES, 256, 0, stream>>>(x, H, lin_w, lin_b, dvis, xs);
  hg_col<<<dim3(N_EDGES / 256, N_NODES / 500), 256, 0, stream>>>(H, De);
  hg_wde<<<N_EDGES / 256, 256, 0, stream>>>(W, De, wde);
  hg_gemm1<<<N_EDGES / 32, 256, 0, stream>>>(H, xs, wde, ts);
  hg_gemm2<<<(N_NODES + 63) / 64, 256, 0, stream>>>(H, ts, dvis, out);
}